// SSLMaskingLayer3D_43490838840027
// MI455X (gfx1250) — compile-verified
//
#include <hip/hip_runtime.h>
#include <cstdint>
#include <cstddef>

// ---- problem constants (from reference setup) ----
#define B_      4
#define HWD_    96            // H = W = D = 96
#define C_      32
#define WINSZ   8
#define NWH     12            // 96/8 windows per axis
#define NUM_WIN 1728          // 12*12*12
#define LEN_KEEP 691          // int(1728 * (1.0 - 0.6)) = int(691.2)
#define BDIM    256

typedef float f4 __attribute__((ext_vector_type(4)));

// ---------------------------------------------------------------------------
// Kernel A: per-batch window mask via rank counting (== stable argsort, keep
// first LEN_KEEP). Noise row staged into LDS with CDNA5 async global->LDS
// copies (ASYNCcnt path), then O(N^2) rank count out of LDS.
// Grid: B_ blocks of BDIM threads.
// ---------------------------------------------------------------------------
__global__ void ssl_win_mask_kernel(const float* __restrict__ noise,
                                    float* __restrict__ win_mask) {
    __shared__ float s_noise[NUM_WIN];
    const int b   = blockIdx.x;
    const int tid = threadIdx.x;
    const float* row = noise + b * NUM_WIN;

    // Async copy of the 1728-float noise row into LDS (gfx1250 ASYNC path).
    for (int i = tid; i < NUM_WIN; i += BDIM) {
        uint32_t lds_addr = (uint32_t)(uintptr_t)(&s_noise[i]);
        uint32_t goff     = (uint32_t)(i * sizeof(float));
        asm volatile("global_load_async_to_lds_b32 %0, %1, %2"
                     :
                     : "v"(lds_addr), "v"(goff), "s"(row)
                     : "memory");
    }
    asm volatile("s_wait_asynccnt 0" ::: "memory");
    __syncthreads();

    // rank(i) = #{j : v_j < v_i} + #{j < i : v_j == v_i}  (stable sort order)
    for (int i = tid; i < NUM_WIN; i += BDIM) {
        const float vi = s_noise[i];
        int rank = 0;
        #pragma unroll 4
        for (int j = 0; j < NUM_WIN; ++j) {
            const float vj = s_noise[j];
            rank += (int)((vj < vi) | ((vj == vi) & (j < i)));
        }
        win_mask[b * NUM_WIN + i] = (rank < LEN_KEEP) ? 0.0f : 1.0f;
    }
}

// ---------------------------------------------------------------------------
// Kernel B: streaming expand + apply.
// Memory layout x[B][96][96][96][32] fp32; innermost D*C row = 3072 floats
// = 12 KB = 12 aligned 1KB window segments. One block = 256 threads * float4
// = 4 KB = exactly 4 consecutive d-windows within ONE (b,h,w) row, so the
// (b,h,w) -> window decomposition is uniform (scalar) per block and each
// thread only picks among 4 mask values via tid>>6.
// All volume traffic uses b128 non-temporal loads/stores (1.36 GB stream,
// never re-read -> don't pollute the 192 MB L2).
// ---------------------------------------------------------------------------
__global__ void ssl_expand_kernel(const f4* __restrict__ x,
                                  const float* __restrict__ win_mask,
                                  f4* __restrict__ x_masked,
                                  f4* __restrict__ mask_out) {
    const int bid = blockIdx.x;
    const int tid = threadIdx.x;

    // Uniform (SGPR) decomposition: bid -> (b,h,w,row-chunk)
    const int row   = bid / 3;              // row index = (b*96 + h)*96 + w
    const int chunk = bid - row * 3;        // which 4-window chunk of the row
    const int w     = row % HWD_;
    const int t2    = row / HWD_;
    const int h     = t2 % HWD_;
    const int b     = t2 / HWD_;
    const int win_base =
        (((b * NWH) + (h >> 3)) * NWH + (w >> 3)) * NWH + chunk * 4;

    // 64 threads (= 1 KB = one 8-voxel d-window * 32 ch) share one mask value.
    const int   win  = win_base + (tid >> 6);
    const float m    = win_mask[win];
    const float keep = 1.0f - m;

    const size_t i4 = (size_t)bid * BDIM + tid;
    f4 v = __builtin_nontemporal_load(x + i4);
    f4 xm = v * keep;
    f4 mv = {m, m, m, m};
    __builtin_nontemporal_store(xm, x_masked + i4);
    __builtin_nontemporal_store(mv, mask_out + i4);
}

// ---------------------------------------------------------------------------
extern "C" void kernel_launch(void* const* d_in, const int* in_sizes, int n_in,
                              void* d_out, int out_size, void* d_ws, size_t ws_size,
                              hipStream_t stream) {
    const float* x     = (const float*)d_in[0];
    const float* noise = (const float*)d_in[1];

    float* win_mask = (float*)d_ws;                 // 4*1728 floats = 27.6 KB
    float* out      = (float*)d_out;

    const size_t nelem = (size_t)B_ * HWD_ * HWD_ * HWD_ * C_;  // 113,246,208
    float* x_masked = out;                          // first output
    float* mask     = out + nelem;                  // second output

    // 1) window-level mask (tiny)
    ssl_win_mask_kernel<<<B_, BDIM, 0, stream>>>(noise, win_mask);

    // 2) streaming expand/apply: nelem/4 float4 elements, 256 per block
    const int nblocks = (int)(nelem / 4 / BDIM);    // 110,592
    ssl_expand_kernel<<<nblocks, BDIM, 0, stream>>>(
        (const f4*)x, win_mask, (f4*)x_masked, (f4*)mask);
}